// SimplicialModel2_1176821040083
// MI455X (gfx1250) — compile-verified
//
#include <hip/hip_runtime.h>
#include <hip/hip_bf16.h>

typedef __attribute__((ext_vector_type(2))) float v2f;
typedef __attribute__((ext_vector_type(4))) float v4f;
typedef __attribute__((ext_vector_type(8))) float v8f;

#define TILE_M 64
#define TILE_N 64
#define TILE_K 16
#define LDS_PAD 2

// ---------------------------------------------------------------------------
// GEMM with bias: C[M,N] = A[M,K] @ W[K,N] + bias[N], fp32 WMMA 16x16x4.
// 128 threads = 4 waves; each wave owns a 16(M) x 64(N) strip of the 64x64 tile.
// Double-buffered LDS; fragments batch-loaded before each 16-WMMA burst.
// ---------------------------------------------------------------------------
__global__ __launch_bounds__(128) void simp_gemm_bias_wmma(
    const float* __restrict__ A, const float* __restrict__ W,
    const float* __restrict__ bias, float* __restrict__ C,
    int M, int K, int N)
{
    __shared__ float As[2][TILE_M][TILE_K + LDS_PAD];   // [m][k]
    __shared__ float Bs[2][TILE_N][TILE_K + LDS_PAD];   // W tile transposed: [n][k]

    const int tid  = threadIdx.x;
    const int wave = tid >> 5;
    const int lane = tid & 31;
    const int half = lane >> 4;    // 0: lanes 0-15, 1: lanes 16-31
    const int lm   = lane & 15;
    const int m0   = blockIdx.y * TILE_M;
    const int n0   = blockIdx.x * TILE_N;

    // A staging: 64x16 floats as 2 float4 per thread (rows ar0, ar0+32).
    const int ar0 = tid >> 2, ac0 = (tid & 3) * 4;
    // W staging: 16x64 floats as 8 scalars per thread (rows wk0+2j, col wn).
    const int wk0 = tid >> 6, wn = tid & 63;

    v8f acc[4];
#pragma unroll
    for (int t = 0; t < 4; ++t)
#pragma unroll
        for (int r = 0; r < 8; ++r) acc[t][r] = 0.0f;

    // ---- prologue: stage tile 0 into buffer 0 ----
    {
        v4f a0 = *(const v4f*)&A[(size_t)(m0 + ar0) * K + ac0];
        v4f a1 = *(const v4f*)&A[(size_t)(m0 + ar0 + 32) * K + ac0];
        float wv[8];
#pragma unroll
        for (int j = 0; j < 8; ++j)
            wv[j] = W[(size_t)(wk0 + 2 * j) * N + (n0 + wn)];
        *(v4f*)&As[0][ar0][ac0]      = a0;
        *(v4f*)&As[0][ar0 + 32][ac0] = a1;
#pragma unroll
        for (int j = 0; j < 8; ++j) Bs[0][wn][wk0 + 2 * j] = wv[j];
    }
    __syncthreads();

    const int nT = K / TILE_K;
    for (int t = 0; t < nT; ++t) {
        const int p = t & 1;
        const bool more = (t + 1 < nT);

        // ---- issue global loads for the next tile (overlap with WMMA) ----
        v4f na0, na1;
        float nw[8];
        if (more) {
            int k0 = (t + 1) * TILE_K;
            na0 = *(const v4f*)&A[(size_t)(m0 + ar0) * K + (k0 + ac0)];
            na1 = *(const v4f*)&A[(size_t)(m0 + ar0 + 32) * K + (k0 + ac0)];
#pragma unroll
            for (int j = 0; j < 8; ++j)
                nw[j] = W[(size_t)(k0 + wk0 + 2 * j) * N + (n0 + wn)];
        }

        // ---- batch-load all fragments for this chunk ----
        const float (*Ap)[TILE_K + LDS_PAD] = As[p];
        const float (*Bp)[TILE_K + LDS_PAD] = Bs[p];
        v2f a[4], b[4][4];
#pragma unroll
        for (int k4 = 0; k4 < 4; ++k4)
            a[k4] = *(const v2f*)&Ap[wave * 16 + lm][4 * k4 + 2 * half];
#pragma unroll
        for (int tt = 0; tt < 4; ++tt)
#pragma unroll
            for (int k4 = 0; k4 < 4; ++k4)
                b[tt][k4] = *(const v2f*)&Bp[tt * 16 + lm][4 * k4 + 2 * half];

        // ---- 16 WMMAs back-to-back ----
#pragma unroll
        for (int k4 = 0; k4 < 4; ++k4)
#pragma unroll
            for (int tt = 0; tt < 4; ++tt)
                acc[tt] = __builtin_amdgcn_wmma_f32_16x16x4_f32(
                    false, a[k4], false, b[tt][k4], (short)0, acc[tt], false, false);

        // ---- commit next tile to the other buffer ----
        if (more) {
            *(v4f*)&As[1 - p][ar0][ac0]      = na0;
            *(v4f*)&As[1 - p][ar0 + 32][ac0] = na1;
#pragma unroll
            for (int j = 0; j < 8; ++j) Bs[1 - p][wn][wk0 + 2 * j] = nw[j];
            __syncthreads();
        }
    }

    // C/D layout: VGPR r -> M = r + 8*half, N = lm within each 16x16 tile.
#pragma unroll
    for (int t = 0; t < 4; ++t) {
#pragma unroll
        for (int r = 0; r < 8; ++r) {
            int row = m0 + wave * 16 + r + 8 * half;
            int col = n0 + t * 16 + lm;
            C[(size_t)row * N + col] = acc[t][r] + bias[col];
        }
    }
}

// ---------------------------------------------------------------------------
// COO scatter SpMM: m[rows[e]+rowOff, f] += vals[e] * h[cols[e]+colOff, f]
// One thread per (nnz, 4 features); fo = 1<<lf, lf4 = lf-2.
// ---------------------------------------------------------------------------
__global__ __launch_bounds__(256) void simp_spmm_scatter(
    const int* __restrict__ rows, const int* __restrict__ cols,
    const float* __restrict__ vals,
    const float* __restrict__ h, float* __restrict__ m,
    int nnz, int lf, int rowOff, int colOff)
{
    const int lf4 = lf - 2;
    long long idx = (long long)blockIdx.x * blockDim.x + threadIdx.x;
    long long total = (long long)nnz << lf4;
    if (idx >= total) return;
    int e  = (int)(idx >> lf4);
    int f4 = (int)(idx & ((1 << lf4) - 1));
    int r = rows[e] + rowOff;
    int c = cols[e] + colOff;
    float v = vals[e];
    v4f hv = *(const v4f*)&h[((size_t)c << lf) + 4 * f4];
    float* dst = &m[((size_t)r << lf) + 4 * f4];
    atomicAdd(dst + 0, v * hv.x);
    atomicAdd(dst + 1, v * hv.y);
    atomicAdd(dst + 2, v * hv.z);
    atomicAdd(dst + 3, v * hv.w);
}

__global__ __launch_bounds__(256) void simp_leaky_relu(float* __restrict__ x, long long n4)
{
    long long i = (long long)blockIdx.x * blockDim.x + threadIdx.x;
    if (i < n4) {
        v4f v = *(const v4f*)&x[4 * i];
        v.x = v.x > 0.0f ? v.x : 0.01f * v.x;
        v.y = v.y > 0.0f ? v.y : 0.01f * v.y;
        v.z = v.z > 0.0f ? v.z : 0.01f * v.z;
        v.w = v.w > 0.0f ? v.w : 0.01f * v.w;
        *(v4f*)&x[4 * i] = v;
    }
}

// fe = tanh(es[order][idx] @ lin1_w + lin1_b), es concatenated [88000,1024]
__global__ __launch_bounds__(256) void simp_head_fe(
    const float* __restrict__ es, const float* __restrict__ lin1_w,
    const float* __restrict__ lin1_b, const int* __restrict__ order_p,
    const int* __restrict__ idx_p, float* __restrict__ fe)
{
    __shared__ float row[1024];
    const int offs[4] = {0, 20000, 60000, 80000};
    int order = order_p[0];
    int idx   = idx_p[0];
    const float* src = es + (size_t)(offs[order & 3] + idx) * 1024;
    for (int k = threadIdx.x; k < 1024; k += 256) row[k] = src[k];
    __syncthreads();
    int j = threadIdx.x;  // 0..255
    float s = lin1_b[j];
    for (int k = 0; k < 1024; ++k) s += row[k] * lin1_w[k * 256 + j];
    fe[j] = tanhf(s);
}

// scores[c] = [fe | rel_embed[c]] @ rel_w + rel_b; gather first 16 nonzeros of rel
__global__ __launch_bounds__(128) void simp_head_scores(
    const float* __restrict__ fe, const float* __restrict__ rel_w,
    const float* __restrict__ rel_b, const float* __restrict__ rel_embed,
    const int* __restrict__ rel, float* __restrict__ out)
{
    __shared__ float scores[128];
    int c = threadIdx.x;  // 0..127
    float s = rel_b[0];
    for (int j = 0; j < 256; ++j) s += fe[j] * rel_w[j];
    for (int j = 0; j < 256; ++j) s += rel_embed[c * 256 + j] * rel_w[256 + j];
    scores[c] = s;
    __syncthreads();
    if (c == 0) {
        int cnt = 0;
        for (int i = 0; i < 128 && cnt < 16; ++i)
            if (rel[i] != 0) out[cnt++] = scores[i];
        for (; cnt < 16; ++cnt) out[cnt] = scores[0];  // jnp.nonzero(size=16) pads with 0
    }
}

static inline void launch_scatter(const int* rows, const int* cols, const float* vals,
                                  const float* h, float* m, int nnz, int lf,
                                  int rowOff, int colOff, hipStream_t stream)
{
    long long total = (long long)nnz << (lf - 2);
    unsigned blocks = (unsigned)((total + 255) / 256);
    simp_spmm_scatter<<<blocks, 256, 0, stream>>>(rows, cols, vals, h, m, nnz, lf, rowOff, colOff);
}

extern "C" void kernel_launch(void* const* d_in, const int* in_sizes, int n_in,
                              void* d_out, int out_size, void* d_ws, size_t ws_size,
                              hipStream_t stream)
{
    (void)in_sizes; (void)n_in; (void)out_size; (void)ws_size;

    const float* W[3]  = {(const float*)d_in[25], (const float*)d_in[27], (const float*)d_in[29]};
    const float* bb[3] = {(const float*)d_in[26], (const float*)d_in[28], (const float*)d_in[30]};
    const int*   Lr[4]; const int* Lc[4]; const float* Lv[4];
    for (int i = 0; i < 4; ++i) {
        Lr[i] = (const int*)d_in[4 + 3 * i];
        Lc[i] = (const int*)d_in[5 + 3 * i];
        Lv[i] = (const float*)d_in[6 + 3 * i];
    }
    const int* Br[3]; const int* Bc[3]; const float* Bv[3];
    for (int j = 0; j < 3; ++j) {
        Br[j] = (const int*)d_in[16 + 3 * j];
        Bc[j] = (const int*)d_in[17 + 3 * j];
        Bv[j] = (const float*)d_in[18 + 3 * j];
    }
    const float* lin1_w    = (const float*)d_in[31];
    const float* lin1_b    = (const float*)d_in[32];
    const float* rel_w     = (const float*)d_in[33];
    const float* rel_b     = (const float*)d_in[34];
    const float* rel_embed = (const float*)d_in[35];
    const int*   order_p   = (const int*)d_in[36];
    const int*   idx_p     = (const int*)d_in[37];
    const int*   rel       = (const int*)d_in[38];

    // Workspace: P0/P1 are 1024-wide, P2 is 512-wide (also holds the 128-wide input concat).
    float* ws = (float*)d_ws;
    float* P0 = ws;
    float* P1 = ws + (size_t)88000 * 1024;
    float* P2 = ws + (size_t)2 * 88000 * 1024;
    float* FE = ws + (size_t)88000 * (1024 + 1024 + 512);

    const int Nr[4]  = {20000, 40000, 20000, 8000};
    const int off[4] = {0, 20000, 60000, 80000};
    const int LN[4]  = {320000, 640000, 320000, 128000};
    const int BNZ[3] = {80000, 60000, 32000};

    // Concatenate e0..e3 into P2 as [88000, 128].
    for (int i = 0; i < 4; ++i)
        hipMemcpyAsync(P2 + (size_t)off[i] * 128, d_in[i],
                       (size_t)Nr[i] * 128 * sizeof(float),
                       hipMemcpyDeviceToDevice, stream);

    // Buffer rotation: L1 es=P2(128) h=P0(256) m=P1; L2 es=P1 h=P0(512) m=P2;
    //                  L3 es=P2(512) h=P1(1024) m=P0.
    const float* esp[3] = {P2, P1, P2};
    float*       hp[3]  = {P0, P0, P1};
    float*       mp[3]  = {P1, P2, P0};
    const int    fi[3]  = {128, 256, 512};
    const int    fo[3]  = {256, 512, 1024};
    const int    lf[3]  = {8, 9, 10};

    for (int l = 0; l < 3; ++l) {
        dim3 gg(fo[l] / TILE_N, 88000 / TILE_M);
        simp_gemm_bias_wmma<<<gg, 128, 0, stream>>>(esp[l], W[l], bb[l], hp[l],
                                                    88000, fi[l], fo[l]);
        hipMemsetAsync(mp[l], 0, (size_t)88000 * fo[l] * sizeof(float), stream);
        for (int i = 0; i < 4; ++i) {
            // L_i @ h_i
            launch_scatter(Lr[i], Lc[i], Lv[i], hp[l], mp[l], LN[i], lf[l],
                           off[i], off[i], stream);
            // B_i^T @ h_{i-1}: scatter rows = B_cols (level i), gather = B_rows (level i-1)
            if (i > 0)
                launch_scatter(Bc[i - 1], Br[i - 1], Bv[i - 1], hp[l], mp[l],
                               BNZ[i - 1], lf[l], off[i], off[i - 1], stream);
            // B_{i+1} @ h_{i+1}: scatter rows = B_rows (level i), gather = B_cols (level i+1)
            if (i < 3)
                launch_scatter(Br[i], Bc[i], Bv[i], hp[l], mp[l],
                               BNZ[i], lf[l], off[i], off[i + 1], stream);
        }
        long long tot4 = 88000ll * fo[l] / 4;
        simp_leaky_relu<<<(unsigned)((tot4 + 255) / 256), 256, 0, stream>>>(mp[l], tot4);
    }

    simp_head_fe<<<1, 256, 0, stream>>>(P0, lin1_w, lin1_b, order_p, idx_p, FE);
    simp_head_scores<<<1, 128, 0, stream>>>(FE, rel_w, rel_b, rel_embed, rel,
                                            (float*)d_out);
}